// GATRegression_22325240004719
// MI455X (gfx1250) — compile-verified
//
#include <hip/hip_runtime.h>

#define NN 100000
#define D 64
#define NEG_SLOPE 0.2f
#define KEY_NEG_INF 0x007FFFFFu  // order-key of -inf

typedef float v2f __attribute__((ext_vector_type(2)));
typedef float v8f __attribute__((ext_vector_type(8)));

// ---- monotone float <-> uint key for atomic max over signed floats ----
__device__ __forceinline__ unsigned f32_key(float f) {
  unsigned u = __float_as_uint(f);
  return (u & 0x80000000u) ? ~u : (u | 0x80000000u);
}
__device__ __forceinline__ float key_f32(unsigned k) {
  unsigned u = (k & 0x80000000u) ? (k ^ 0x80000000u) : ~k;
  return __uint_as_float(u);
}

// ---------------- generic fill ----------------
__global__ void fill_u32(unsigned* __restrict__ p, unsigned v, int n) {
  int i = blockIdx.x * blockDim.x + threadIdx.x;
  if (i < n) p[i] = v;
}

// ---------------- WMMA fp32 GEMM: H[N,64] = X[N,64] @ W[64,64] ----------------
// grid.x = N/16 (M tiles); block = 128 threads = 4 waves; wave w owns N-tile w.
// V_WMMA_F32_16X16X4_F32 layouts (wave32):
//   A 16x4 : lane m(0..15)/m+16 holds K={0,1}/{2,3} in 2 VGPRs
//   B 4x16 : vgpr r holds rows K=r (lanes 0-15) and K=r+2 (lanes 16-31)
//   C 16x16: vgpr r holds row M=r (lanes 0-15) and M=r+8 (lanes 16-31)
__global__ void __launch_bounds__(128) gemm16(const float* __restrict__ X,
                                              const float* __restrict__ W,
                                              float* __restrict__ H) {
  const int lane  = threadIdx.x & 31;
  const int wave  = threadIdx.x >> 5;   // N tile 0..3
  const int mtile = blockIdx.x;
  const int m     = lane & 15;
  const int hi    = lane >> 4;          // 0 or 1
  const int ncol  = wave * 16 + m;

  const float* xrow = X + (size_t)(mtile * 16 + m) * D;
  v8f c = {};
#pragma unroll
  for (int k0 = 0; k0 < D; k0 += 4) {
    v2f a, b;
    const int ka = k0 + 2 * hi;
    a.x = xrow[ka];
    a.y = xrow[ka + 1];
    b.x = W[(size_t)ka * D + ncol];
    b.y = W[(size_t)(ka + 1) * D + ncol];
    c = __builtin_amdgcn_wmma_f32_16x16x4_f32(false, a, false, b,
                                              (short)0, c, false, false);
  }
  float* hout = H + (size_t)(mtile * 16 + hi * 8) * D + ncol;
#pragma unroll
  for (int r = 0; r < 8; ++r) hout[(size_t)r * D] = c[r];
}

// ---------------- per-node attention logits: es = h.a_src, ed = h.a_dst ----------------
__global__ void node_attn(const float* __restrict__ H, const float* __restrict__ a_s,
                          const float* __restrict__ a_d, float* __restrict__ es,
                          float* __restrict__ ed, int n) {
  int i = blockIdx.x * blockDim.x + threadIdx.x;
  if (i >= n) return;
  const float* h = H + (size_t)i * D;
  float s = 0.f, d = 0.f;
#pragma unroll
  for (int k = 0; k < D; ++k) { float v = h[k]; s += v * a_s[k]; d += v * a_d[k]; }
  es[i] = s;
  ed[i] = d;
}

// ---------------- pass 1: edge score + segment max over dst ----------------
__global__ void edge_max(const int* __restrict__ src, const int* __restrict__ dst,
                         const float* __restrict__ es, const float* __restrict__ ed,
                         float* __restrict__ escore, unsigned* __restrict__ mkey,
                         int E, int Etot) {
  int e = blockIdx.x * blockDim.x + threadIdx.x;
  if (e >= Etot) return;
  int s, d;
  if (e < E) { s = src[e]; d = dst[e]; } else { s = e - E; d = s; }
  float v = es[s] + ed[d];
  v = (v > 0.f) ? v : NEG_SLOPE * v;   // leaky relu
  escore[e] = v;
  atomicMax(&mkey[d], f32_key(v));
}

// ---------------- pass 2: exp(e - m[dst]) + segment sum ----------------
__global__ void edge_exp(const int* __restrict__ dst, const float* __restrict__ escore,
                         const unsigned* __restrict__ mkey, float* __restrict__ eexp,
                         float* __restrict__ denom, int E, int Etot) {
  int e = blockIdx.x * blockDim.x + threadIdx.x;
  if (e >= Etot) return;
  int d = (e < E) ? dst[e] : (e - E);
  float ex = __expf(escore[e] - key_f32(mkey[d]));
  eexp[e] = ex;
  atomicAdd(&denom[d], ex);
}

// ---------------- pass 3: out[dst] += alpha * h[src]  (16 threads/edge, float4) ----------------
__global__ void edge_scatter(const int* __restrict__ src, const int* __restrict__ dst,
                             const float* __restrict__ H, const float* __restrict__ eexp,
                             const float* __restrict__ denom, float* __restrict__ out,
                             int E, int Etot) {
  long long idx = (long long)blockIdx.x * blockDim.x + threadIdx.x;
  int e    = (int)(idx >> 4);
  int part = (int)(idx & 15);
  if (e >= Etot) return;
  int s, d;
  if (e < E) { s = src[e]; d = dst[e]; } else { s = e - E; d = s; }
  float alpha = eexp[e] / (denom[d] + 1e-16f);
  const float4 hv = *reinterpret_cast<const float4*>(H + (size_t)s * D + part * 4);
  float* o = out + (size_t)d * D + part * 4;
  atomicAdd(o + 0, alpha * hv.x);
  atomicAdd(o + 1, alpha * hv.y);
  atomicAdd(o + 2, alpha * hv.z);
  atomicAdd(o + 3, alpha * hv.w);
}

// ---------------- bias + relu (in place) ----------------
__global__ void bias_relu(float* __restrict__ h, const float* __restrict__ b, int total) {
  int i = blockIdx.x * blockDim.x + threadIdx.x;
  if (i >= total) return;
  float v = h[i] + b[i & (D - 1)];
  h[i] = v > 0.f ? v : 0.f;
}

// ---------------- column sums for mean pooling ----------------
__global__ void colsum(const float* __restrict__ H, float* __restrict__ gsum, int n) {
  int col = threadIdx.x;  // blockDim.x == 64
  float acc = 0.f;
  for (int r = blockIdx.x; r < n; r += gridDim.x)
    acc += H[(size_t)r * D + col];
  atomicAdd(&gsum[col], acc);
}

// ---------------- final: out = (gsum/N) . W_out + b_out ----------------
__global__ void final_out(const float* __restrict__ gsum, const float* __restrict__ Wout,
                          const float* __restrict__ bout, float* __restrict__ out,
                          float invN) {
  __shared__ float red[D];
  int t = threadIdx.x;  // 64 threads
  red[t] = gsum[t] * invN * Wout[t];
  __syncthreads();
  for (int s = 32; s > 0; s >>= 1) {
    if (t < s) red[t] += red[t + s];
    __syncthreads();
  }
  if (t == 0) out[0] = red[0] + bout[0];
}

// ---------------- one GAT layer ----------------
static void run_layer(const float* Xin, const float* W, const float* a_s,
                      const float* a_d, const float* b, float* Hlin, float* Hout,
                      const int* src, const int* dst, float* es, float* ed,
                      unsigned* mkey, float* denom, float* escore, float* eexp,
                      int E, int Etot, hipStream_t stream) {
  const int N = NN;
  gemm16<<<N / 16, 128, 0, stream>>>(Xin, W, Hlin);
  node_attn<<<(N + 255) / 256, 256, 0, stream>>>(Hlin, a_s, a_d, es, ed, N);
  fill_u32<<<(N + 255) / 256, 256, 0, stream>>>(mkey, KEY_NEG_INF, N);
  fill_u32<<<(N + 255) / 256, 256, 0, stream>>>((unsigned*)denom, 0u, N);
  fill_u32<<<(N * D + 255) / 256, 256, 0, stream>>>((unsigned*)Hout, 0u, N * D);
  edge_max<<<(Etot + 255) / 256, 256, 0, stream>>>(src, dst, es, ed, escore, mkey, E, Etot);
  edge_exp<<<(Etot + 255) / 256, 256, 0, stream>>>(dst, escore, mkey, eexp, denom, E, Etot);
  long long sth = (long long)Etot * 16;
  edge_scatter<<<(unsigned)((sth + 255) / 256), 256, 0, stream>>>(src, dst, Hlin, eexp,
                                                                  denom, Hout, E, Etot);
  bias_relu<<<(N * D + 255) / 256, 256, 0, stream>>>(Hout, b, N * D);
}

extern "C" void kernel_launch(void* const* d_in, const int* in_sizes, int n_in,
                              void* d_out, int out_size, void* d_ws, size_t ws_size,
                              hipStream_t stream) {
  const float* x    = (const float*)d_in[0];
  const int*   eidx = (const int*)d_in[1];
  const float* W1   = (const float*)d_in[2];
  const float* as1  = (const float*)d_in[3];
  const float* ad1  = (const float*)d_in[4];
  const float* b1   = (const float*)d_in[5];
  const float* W2   = (const float*)d_in[6];
  const float* as2  = (const float*)d_in[7];
  const float* ad2  = (const float*)d_in[8];
  const float* b2   = (const float*)d_in[9];
  const float* Wout = (const float*)d_in[10];
  const float* bout = (const float*)d_in[11];
  float* out = (float*)d_out;

  const int N = NN;
  const int E = in_sizes[1] / 2;
  const int Etot = E + N;
  const int* src = eidx;
  const int* dst = eidx + E;

  // workspace carve-up (~64 MB total)
  float*    hA     = (float*)d_ws;          // N*D
  float*    hB     = hA + (size_t)N * D;    // N*D
  float*    es     = hB + (size_t)N * D;    // N
  float*    ed     = es + N;                // N
  unsigned* mkey   = (unsigned*)(ed + N);   // N
  float*    denom  = (float*)(mkey + N);    // N
  float*    escore = denom + N;             // Etot
  float*    eexp   = escore + Etot;         // Etot
  float*    gsum   = eexp + Etot;           // 64

  // layer 1: x -> hA (linear) -> hB (aggregated, +b1, relu)
  run_layer(x, W1, as1, ad1, b1, hA, hB, src, dst, es, ed, mkey, denom, escore, eexp, E, Etot, stream);
  // layer 2: hB -> hA (linear) -> hB (aggregated, +b2, relu); gemm consumes hB before it is zeroed
  run_layer(hB, W2, as2, ad2, b2, hA, hB, src, dst, es, ed, mkey, denom, escore, eexp, E, Etot, stream);

  // mean pool + linear head
  fill_u32<<<1, 64, 0, stream>>>((unsigned*)gsum, 0u, D);
  colsum<<<256, 64, 0, stream>>>(hB, gsum, N);
  final_out<<<1, 64, 0, stream>>>(gsum, Wout, bout, out, 1.0f / (float)N);
}